// BindingSiteGraphSAGE_49735721288424
// MI455X (gfx1250) — compile-verified
//
#include <hip/hip_runtime.h>
#include <hip/hip_bf16.h>

#define N_NODES 50000
#define N_EDGES 800000
#define SLOPE   0.15f
#define MT      5          // M-tiles per wave: 50000/16 = 3125 = 5*625

typedef __attribute__((ext_vector_type(2))) float v2f;
typedef __attribute__((ext_vector_type(8))) float v8f;

// ---------------------------------------------------------------------------
// Degree count: cnt[dst] += 1 for every edge (counts identical for all layers)
// ---------------------------------------------------------------------------
__global__ void count_deg_kernel(const int* __restrict__ dst, float* __restrict__ cnt) {
    int e = blockIdx.x * blockDim.x + threadIdx.x;
    if (e < N_EDGES) unsafeAtomicAdd(&cnt[dst[e]], 1.0f);
}

__global__ void make_inv_kernel(const float* __restrict__ cnt, float* __restrict__ inv) {
    int i = blockIdx.x * blockDim.x + threadIdx.x;
    if (i < N_NODES) inv[i] = 1.0f / fmaxf(cnt[i], 1.0f);
}

// ---------------------------------------------------------------------------
// Edge scatter of the *mean*: out[dst] += h[src] * inv[dst].
// One wave per edge, float4 gathers, native f32 global atomics.
// Folding inv[dst] here removes the row-scale multiply from the GEMM hot loop
// (diag(inv)*S*W == (diag(inv)*S)*W).
// ---------------------------------------------------------------------------
__global__ void scatter_mean_kernel(const float* __restrict__ h,
                                    const int* __restrict__ src,
                                    const int* __restrict__ dst,
                                    const float* __restrict__ inv,
                                    float* __restrict__ out, int D) {
    int e = blockIdx.x * (blockDim.x >> 5) + (threadIdx.x >> 5);
    if (e >= N_EDGES) return;
    int lane = threadIdx.x & 31;
    int s = src[e];
    int d = dst[e];
    float sc = inv[d];                    // wave-uniform
    const float4* hs = (const float4*)(h + (size_t)s * D);
    float* od = out + (size_t)d * D;
    int nq = D >> 2;                      // float4 chunks per row (D multiple of 128)
    for (int q = lane; q < nq; q += 32) {
        float4 v = hs[q];
        int j = q << 2;
        unsafeAtomicAdd(&od[j + 0], v.x * sc);
        unsafeAtomicAdd(&od[j + 1], v.y * sc);
        unsafeAtomicAdd(&od[j + 2], v.z * sc);
        unsafeAtomicAdd(&od[j + 3], v.w * sc);
    }
}

// ---------------------------------------------------------------------------
// One GEMM operand pass: c[t] += A@B for MT 16-row tiles, K stepped by 4.
// Load phase (5x A b64 + 2x B b32) is separated from the 5 WMMAs so the
// compiler can batch loads and use partial s_wait_loadcnt values.
// B row pointers advance by 4*N per step (no per-step 64-bit multiplies).
// ---------------------------------------------------------------------------
__device__ __forceinline__ void mma_pass(const float* __restrict__ A,
                                         const float* __restrict__ B,
                                         int K, int N, int tileM0,
                                         int lo, int hi, int ncl, float wm,
                                         v8f (&c)[MT]) {
    const float* arow[MT];
#pragma unroll
    for (int t = 0; t < MT; ++t)
        arow[t] = A + (size_t)((tileM0 + t) * 16 + lo) * K + 2 * hi;
    const float* b0p = B + (size_t)(2 * hi) * N + ncl;
    const float* b1p = b0p + N;
    size_t bstep = (size_t)4 * N;
#pragma unroll 4
    for (int k0 = 0; k0 < K; k0 += 4) {
        // ---- load phase ----
        float b0 = *b0p;
        float b1v = *b1p;
        float2 av[MT];
#pragma unroll
        for (int t = 0; t < MT; ++t)
            av[t] = *(const float2*)(arow[t] + k0);   // 8B-aligned b64 load
        b0p += bstep;
        b1p += bstep;
        // ---- math phase ----
        v2f b; b.x = b0 * wm; b.y = b1v * wm;
#pragma unroll
        for (int t = 0; t < MT; ++t) {
            v2f a; a.x = av[t].x; a.y = av[t].y;
            c[t] = __builtin_amdgcn_wmma_f32_16x16x4_f32(false, a, false, b,
                                                         (short)0, c[t], false, false);
        }
    }
}

// ---------------------------------------------------------------------------
// Fused SAGE GEMM on V_WMMA_F32_16X16X4_F32, wave32.
//   out = act( A1 @ B1 + A2 @ B2 + bias )        (A1 already holds the mean)
// Each wave owns one 16-wide N tile and MT consecutive 16-row M tiles.
// GUARD=false: N multiple of 16 -> unconditional loads, no exec juggling.
// GUARD=true : clamp column index + 0/1 mask multiply (branch-free loads).
// ISA layouts: A lane L holds A[L%16][k0 + 2*(L>=16) + v]   (float2/lane/step)
//              B lane L holds B[k0 + 2*(L>=16) + v][tileN*16 + L%16]
//              C/D elem v  ->  row tileM*16 + v + 8*(L>=16), col tileN*16 + L%16
// ---------------------------------------------------------------------------
template <bool LEAKY, bool GUARD>
__global__ void wmma_gemm_kernel(const float* __restrict__ A1,
                                 const float* __restrict__ B1,
                                 const float* __restrict__ A2,
                                 const float* __restrict__ B2,
                                 const float* __restrict__ bias,
                                 float* __restrict__ out,
                                 int K1, int K2, int N) {
    int wave  = threadIdx.x >> 5;
    int tileN = blockIdx.x * (blockDim.x >> 5) + wave;
    if (tileN * 16 >= N) return;                    // wave-uniform exit
    int tileM0 = blockIdx.y * MT;
    int lane   = threadIdx.x & 31;
    int hi     = lane >> 4;                         // lane half selects K pair
    int lo     = lane & 15;
    int n      = tileN * 16 + lo;                   // logical output column
    bool nok   = !GUARD || (n < N);
    int  ncl   = GUARD ? (nok ? n : 0) : n;         // clamped (always in-bounds)
    float wm   = GUARD ? (nok ? 1.0f : 0.0f) : 1.0f;

    v8f c[MT];
#pragma unroll
    for (int t = 0; t < MT; ++t)
        c[t] = (v8f){0.f, 0.f, 0.f, 0.f, 0.f, 0.f, 0.f, 0.f};

    mma_pass(A1, B1, K1, N, tileM0, lo, hi, ncl, wm, c);   // mean x Wl
    if (A2)
        mma_pass(A2, B2, K2, N, tileM0, lo, hi, ncl, wm, c); // root x Wr

    float bb = bias[ncl] * wm;
#pragma unroll
    for (int t = 0; t < MT; ++t) {
#pragma unroll
        for (int v = 0; v < 8; ++v) {
            int r = (tileM0 + t) * 16 + v + 8 * hi;
            float val = c[t][v] + bb;
            if (LEAKY) val = (val > 0.0f) ? val : SLOPE * val;
            if (nok) out[(size_t)r * N + n] = val;
        }
    }
}

static void gemm_launch(bool leaky,
                        const float* A1, const float* B1,
                        const float* A2, const float* B2,
                        const float* bias, float* out,
                        int K1, int K2, int N, hipStream_t stream) {
    int ntiles = (N + 15) / 16;
    dim3 grid((ntiles + 3) / 4, (N_NODES / 16) / MT);   // 4 waves (N-tiles) per block
    dim3 block(128);
    bool guard = (N % 16) != 0;
    if (guard) {
        if (leaky)
            wmma_gemm_kernel<true, true><<<grid, block, 0, stream>>>(A1, B1, A2, B2, bias, out, K1, K2, N);
        else
            wmma_gemm_kernel<false, true><<<grid, block, 0, stream>>>(A1, B1, A2, B2, bias, out, K1, K2, N);
    } else {
        if (leaky)
            wmma_gemm_kernel<true, false><<<grid, block, 0, stream>>>(A1, B1, A2, B2, bias, out, K1, K2, N);
        else
            wmma_gemm_kernel<false, false><<<grid, block, 0, stream>>>(A1, B1, A2, B2, bias, out, K1, K2, N);
    }
}

// ---------------------------------------------------------------------------
// Host orchestration
// ---------------------------------------------------------------------------
extern "C" void kernel_launch(void* const* d_in, const int* in_sizes, int n_in,
                              void* d_out, int out_size, void* d_ws, size_t ws_size,
                              hipStream_t stream) {
    const float* x   = (const float*)d_in[0];
    const int*   ei  = (const int*)d_in[1];
    const int*   src = ei;                 // edge_index[0]
    const int*   dst = ei + N_EDGES;       // edge_index[1]
    // d_in[2..4] = edge_attr / We / be -> dead code in reference, skipped
    const float* W1l = (const float*)d_in[5];
    const float* b1  = (const float*)d_in[6];
    const float* W1r = (const float*)d_in[7];
    const float* W2l = (const float*)d_in[8];
    const float* b2  = (const float*)d_in[9];
    const float* W2r = (const float*)d_in[10];
    const float* W3l = (const float*)d_in[11];
    const float* b3  = (const float*)d_in[12];
    const float* W3r = (const float*)d_in[13];
    const float* Wp  = (const float*)d_in[14];
    const float* bp  = (const float*)d_in[15];
    const float* Wf1 = (const float*)d_in[16];
    const float* bf1 = (const float*)d_in[17];
    const float* Wf2 = (const float*)d_in[18];
    const float* bf2 = (const float*)d_in[19];

    // workspace layout (floats), all 16B-aligned
    float* ws   = (float*)d_ws;
    float* cnt  = ws;                                  // 50000
    float* inv  = cnt + N_NODES;                       // 50000
    float* bufA = inv + N_NODES;                       // 50000*512 (mean buffer)
    float* bufB = bufA + (size_t)N_NODES * 512;        // 50000*512 (h1 / h3)
    float* bufC = bufB + (size_t)N_NODES * 512;        // 50000*256 (h2 / h4)

    // --- degrees (shared by all layers) ---
    hipMemsetAsync(cnt, 0, N_NODES * sizeof(float), stream);
    count_deg_kernel<<<(N_EDGES + 255) / 256, 256, 0, stream>>>(dst, cnt);
    make_inv_kernel<<<(N_NODES + 255) / 256, 256, 0, stream>>>(cnt, inv);

    // --- layer 1: 128 -> 512 ---
    hipMemsetAsync(bufA, 0, (size_t)N_NODES * 128 * sizeof(float), stream);
    scatter_mean_kernel<<<(N_EDGES + 7) / 8, 256, 0, stream>>>(x, src, dst, inv, bufA, 128);
    gemm_launch(true, bufA, W1l, x, W1r, b1, bufB, 128, 128, 512, stream);

    // --- layer 2: 512 -> 256 ---
    hipMemsetAsync(bufA, 0, (size_t)N_NODES * 512 * sizeof(float), stream);
    scatter_mean_kernel<<<(N_EDGES + 7) / 8, 256, 0, stream>>>(bufB, src, dst, inv, bufA, 512);
    gemm_launch(true, bufA, W2l, bufB, W2r, b2, bufC, 512, 512, 256, stream);

    // --- layer 3: 256 -> 64 ---
    hipMemsetAsync(bufA, 0, (size_t)N_NODES * 256 * sizeof(float), stream);
    scatter_mean_kernel<<<(N_EDGES + 7) / 8, 256, 0, stream>>>(bufC, src, dst, inv, bufA, 256);
    gemm_launch(true, bufA, W3l, bufC, W3r, b3, bufB, 256, 256, 64, stream);

    // --- head: pre_fc (64->32, no act), fc1 (32->32, leaky), fc2 (32->2) ---
    gemm_launch(false, bufB, Wp,  nullptr, nullptr, bp,  bufC, 64, 0, 32, stream);
    gemm_launch(true,  bufC, Wf1, nullptr, nullptr, bf1, bufA, 32, 0, 32, stream);
    gemm_launch(false, bufA, Wf2, nullptr, nullptr, bf2, (float*)d_out, 32, 0, 2, stream);
}